// GCN_67095979098871
// MI455X (gfx1250) — compile-verified
//
#include <hip/hip_runtime.h>

// GCN 2-layer forward for MI455X (gfx1250, wave32).
// Memory-bound on edge gather/scatter (~1.3 GB traffic -> ~60us @ 23.3 TB/s);
// GEMMs (~2 GFLOP) done with V_WMMA_F32_16X16X4_F32 (exact f32 matrix path).

#define IN_CH  128
#define HID_CH 64
#define OUT_CH 32

typedef __attribute__((ext_vector_type(2))) float v2f;
typedef __attribute__((ext_vector_type(8))) float v8f;

// ---------------- degree kernels ----------------
__global__ void k_deg_init(float* __restrict__ deg, int n) {
    int i = blockIdx.x * blockDim.x + threadIdx.x;
    if (i < n) deg[i] = 1.0f;                    // self-loop contributes 1
}

__global__ void k_deg_accum(const int* __restrict__ da, float* __restrict__ deg, int e) {
    int i = blockIdx.x * blockDim.x + threadIdx.x;
    if (i < e) atomicAdd(&deg[da[i]], 1.0f);
}

// ---------------- self-loop init: dst[i][c] = src[i][c] / deg[i] ----------------
__global__ void k_self_init(const float* __restrict__ src, const float* __restrict__ deg,
                            float* __restrict__ dst, int total, int shift) {
    int i = blockIdx.x * blockDim.x + threadIdx.x;
    if (i < total) {
        float d = deg[i >> shift];               // deg >= 1 always
        dst[i] = src[i] / d;                     // norm(self) = rsqrt(d*d) = 1/d
    }
}

// ---------------- edge scatter: one wave32 per edge, channels across lanes ----------------
template <int C>
__global__ __launch_bounds__(256) void k_scatter(const int* __restrict__ sa,
                                                 const int* __restrict__ da,
                                                 const float* __restrict__ deg,
                                                 const float* __restrict__ src,
                                                 float* __restrict__ dst, int e) {
    int gid  = blockIdx.x * blockDim.x + threadIdx.x;
    int edge = gid >> 5;
    int lane = gid & 31;
    if (edge >= e) return;
    int s = sa[edge];
    int d = da[edge];
    float p    = deg[s] * deg[d];
    float norm = (p > 0.0f) ? rsqrtf(p) : 0.0f;
    if (C == 64) {
        const float2* sp = (const float2*)(src + (size_t)s * 64) + lane;   // 8B/lane, coalesced
        float2 v = *sp;
        float* dp = dst + (size_t)d * 64 + lane * 2;
        atomicAdd(dp,     v.x * norm);           // global_atomic_add_f32 (no return)
        atomicAdd(dp + 1, v.y * norm);
    } else {
        float v = src[(size_t)s * 32 + lane];
        atomicAdd(dst + (size_t)d * 32 + lane, v * norm);
    }
}

// ---------------- WMMA GEMM: Y[n,NC] = (RELU?max(A,0):A)[n,K] @ W[K,NC] + bias ----------------
// One wave per 16x16 output tile; W staged in LDS and shared by the block's 8 waves.
template <int K, int NC, bool RELU>
__global__ __launch_bounds__(256) void k_gemm(const float* __restrict__ A,
                                              const float* __restrict__ W,
                                              const float* __restrict__ bias,
                                              float* __restrict__ Y, int n) {
    __shared__ float Wl[K * NC];
    for (int i = threadIdx.x; i < K * NC; i += 256) Wl[i] = W[i];
    __syncthreads();

    const int colTiles = NC / 16;
    int wave    = (blockIdx.x * 256 + (int)threadIdx.x) >> 5;
    int lane    = threadIdx.x & 31;
    int rowTile = wave / colTiles;
    int colTile = wave % colTiles;
    if (rowTile >= n / 16) return;               // uniform per-wave exit: EXEC stays all-1s

    int row0 = rowTile * 16, col0 = colTile * 16;
    int half = lane >> 4;                        // 0: lanes 0-15, 1: lanes 16-31
    int l    = lane & 15;

    // A 16x4 f32 tile layout (ISA 7.12.2): lane half selects K pair {0,1} vs {2,3}
    const float* ap = A + (size_t)(row0 + l) * K + half * 2;

    v8f acc = {};
#pragma unroll
    for (int k = 0; k < K; k += 4) {
        v2f a;
        a.x = ap[k];
        a.y = ap[k + 1];
        if (RELU) { a.x = fmaxf(a.x, 0.0f); a.y = fmaxf(a.y, 0.0f); }
        // B 4x16 tile: vgpr0 = row (k + 2*half), vgpr1 = row (k + 2*half + 1)
        const float* bp = &Wl[(k + half * 2) * NC + col0 + l];
        v2f b;
        b.x = bp[0];
        b.y = bp[NC];
        acc = __builtin_amdgcn_wmma_f32_16x16x4_f32(false, a, false, b,
                                                    (short)0, acc, false, false);
    }

    // C/D layout: vgpr i -> row (row0 + half*8 + i), col (col0 + l)
    float bv = bias[col0 + l];
    float* yp = Y + (size_t)(row0 + half * 8) * NC + col0 + l;
#pragma unroll
    for (int i = 0; i < 8; ++i) yp[(size_t)i * NC] = acc[i] + bv;
}

extern "C" void kernel_launch(void* const* d_in, const int* in_sizes, int n_in,
                              void* d_out, int out_size, void* d_ws, size_t ws_size,
                              hipStream_t stream) {
    const float* x  = (const float*)d_in[0];
    const int*   ei = (const int*)  d_in[1];
    const float* W1 = (const float*)d_in[2];
    const float* b1 = (const float*)d_in[3];
    const float* W2 = (const float*)d_in[4];
    const float* b2 = (const float*)d_in[5];

    int n = in_sizes[0] / IN_CH;                 // 100000
    int e = in_sizes[1] / 2;                     // 1600000
    const int* sa = ei;                          // ei[0]
    const int* da = ei + e;                      // ei[1]

    // carve scratch from d_ws (256B-aligned blocks): deg | h0 | agg1 | h2  (~64.5 MB)
    char*  ws  = (char*)d_ws;
    size_t off = 0;
    auto carve = [&](size_t bytes) -> void* {
        void* p = ws + off;
        off = (off + bytes + 255) & ~(size_t)255;
        return p;
    };
    float* deg  = (float*)carve((size_t)n * 4);
    float* h0   = (float*)carve((size_t)n * HID_CH * 4);
    float* agg1 = (float*)carve((size_t)n * HID_CH * 4);
    float* h2   = (float*)carve((size_t)n * OUT_CH * 4);
    float* out  = (float*)d_out;

    // 1) degrees (with self loops)
    k_deg_init <<<(unsigned)((n + 255) / 256), 256, 0, stream>>>(deg, n);
    k_deg_accum<<<(unsigned)((e + 255) / 256), 256, 0, stream>>>(da, deg, e);

    // 2) h0 = x @ W1 + b1   (WMMA f32)
    {
        int tiles = (n / 16) * (HID_CH / 16);
        k_gemm<IN_CH, HID_CH, false><<<(unsigned)((tiles + 7) / 8), 256, 0, stream>>>(x, W1, b1, h0, n);
    }

    // 3) layer-1 aggregate: agg1 = selfnorm(h0) + scatter_add(edges)
    k_self_init<<<(unsigned)(((size_t)n * HID_CH + 255) / 256), 256, 0, stream>>>(h0, deg, agg1, n * HID_CH, 6);
    k_scatter<HID_CH><<<(unsigned)(((size_t)e * 32 + 255) / 256), 256, 0, stream>>>(sa, da, deg, h0, agg1, e);

    // 4) h2 = relu(agg1) @ W2 + b2   (WMMA f32, relu fused on A input)
    {
        int tiles = (n / 16) * (OUT_CH / 16);
        k_gemm<HID_CH, OUT_CH, true><<<(unsigned)((tiles + 7) / 8), 256, 0, stream>>>(agg1, W2, b2, h2, n);
    }

    // 5) layer-2 aggregate straight into d_out
    k_self_init<<<(unsigned)(((size_t)n * OUT_CH + 255) / 256), 256, 0, stream>>>(h2, deg, out, n * OUT_CH, 5);
    k_scatter<OUT_CH><<<(unsigned)(((size_t)e * 32 + 255) / 256), 256, 0, stream>>>(sa, da, deg, h2, out, e);
}